// UpInterpoaltion_knn_37855841747272
// MI455X (gfx1250) — compile-verified
//
#include <hip/hip_runtime.h>

typedef __attribute__((ext_vector_type(2))) float v2f;
typedef __attribute__((ext_vector_type(8))) float v8f;

#define Bb 4
#define Ss 4096
#define Nn 16384
#define Cc 256

// ---------------------------------------------------------------------------
// Kernel 1: 3-NN inverse-distance interpolation.
// Block = 256 threads (8 waves), 64 queries/block, xyz1 staged in LDS (48KB).
// ---------------------------------------------------------------------------
__global__ __launch_bounds__(256) void knn_interp_kernel(
    const float* __restrict__ feature_1,   // (B,S,C)
    const float* __restrict__ points_1,    // (B,S,6)
    const float* __restrict__ points_2,    // (B,N,6)
    float* __restrict__ nf)                // (B,N,C) workspace
{
  __shared__ float sx[Ss], sy[Ss], sz[Ss];
  const int tid  = threadIdx.x;
  const int n0   = blockIdx.x * 64;      // 64 queries per block; N%64==0 so one batch
  const int b    = n0 / Nn;
  const float* p1 = points_1 + (size_t)b * Ss * 6;
  for (int i = tid; i < Ss; i += 256) {
    sx[i] = p1[i * 6 + 0];
    sy[i] = p1[i * 6 + 1];
    sz[i] = p1[i * 6 + 2];
  }
  __syncthreads();

  const int lane = tid & 31;
  const int w    = tid >> 5;
  const float* f1b = feature_1 + (size_t)b * Ss * Cc;

  for (int t = 0; t < 8; ++t) {
    const int ng = n0 + w * 8 + t;              // global row in [0, B*N)
    const float* q = points_2 + (size_t)ng * 6; // (b*N + n)*6
    const float qx = q[0], qy = q[1], qz = q[2];

    // lane-local sorted top-3 (ascending d)
    float d0 = 3.4e38f, d1 = 3.4e38f, d2 = 3.4e38f;
    int   i0 = 0,       i1 = 0,       i2 = 0;
    for (int s = lane; s < Ss; s += 32) {
      const float dx = sx[s] - qx, dy = sy[s] - qy, dz = sz[s] - qz;
      const float d  = dx * dx + dy * dy + dz * dz;
      if (d < d0)      { d2 = d1; i2 = i1; d1 = d0; i1 = i0; d0 = d; i0 = s; }
      else if (d < d1) { d2 = d1; i2 = i1; d1 = d;  i1 = s; }
      else if (d < d2) { d2 = d;  i2 = s; }
    }

    // wave32 butterfly reduction; branch-free merge of two sorted triples
    #pragma unroll
    for (int off = 16; off >= 1; off >>= 1) {
      const float o0 = __shfl_xor(d0, off, 32);
      const float o1 = __shfl_xor(d1, off, 32);
      const float o2 = __shfl_xor(d2, off, 32);
      const int   j0 = __shfl_xor(i0, off, 32);
      const int   j1 = __shfl_xor(i1, off, 32);
      const int   j2 = __shfl_xor(i2, off, 32);

      const bool tb0 = o0 < d0;
      const float m0 = tb0 ? o0 : d0;  const int k0i = tb0 ? j0 : i0;
      const float A0 = tb0 ? d0 : d1;  const int A0i = tb0 ? i0 : i1;
      const float A1 = tb0 ? d1 : d2;  const int A1i = tb0 ? i1 : i2;
      const float B0 = tb0 ? o1 : o0;  const int B0i = tb0 ? j1 : j0;
      const float B1 = tb0 ? o2 : o1;  const int B1i = tb0 ? j2 : j1;

      const bool tb1 = B0 < A0;
      const float m1  = tb1 ? B0 : A0;  const int k1i  = tb1 ? B0i : A0i;
      const float A0b = tb1 ? A0 : A1;  const int A0bi = tb1 ? A0i : A1i;
      const float B0b = tb1 ? B1 : B0;  const int B0bi = tb1 ? B1i : B0i;

      const bool tb2 = B0b < A0b;
      const float m2 = tb2 ? B0b : A0b; const int k2i = tb2 ? B0bi : A0bi;

      d0 = m0; d1 = m1; d2 = m2; i0 = k0i; i1 = k1i; i2 = k2i;
    }

    // inverse-distance weights (clamped like reference)
    float w0 = 1.0f / fmaxf(d0, 1e-16f);
    float w1 = 1.0f / fmaxf(d1, 1e-16f);
    float w2 = 1.0f / fmaxf(d2, 1e-16f);
    const float inv = 1.0f / (w0 + w1 + w2);
    w0 *= inv; w1 *= inv; w2 *= inv;

    // weighted gather: lane handles 8 consecutive channels (2 x float4)
    const float* r0 = f1b + (size_t)i0 * Cc + lane * 8;
    const float* r1 = f1b + (size_t)i1 * Cc + lane * 8;
    const float* r2 = f1b + (size_t)i2 * Cc + lane * 8;
    float* o = nf + (size_t)ng * Cc + lane * 8;
    #pragma unroll
    for (int h = 0; h < 2; ++h) {
      const float4 a = *(const float4*)(r0 + h * 4);
      const float4 c = *(const float4*)(r1 + h * 4);
      const float4 e = *(const float4*)(r2 + h * 4);
      float4 v;
      v.x = w0 * a.x + w1 * c.x + w2 * e.x;
      v.y = w0 * a.y + w1 * c.y + w2 * e.y;
      v.z = w0 * a.z + w1 * c.z + w2 * e.z;
      v.w = w0 * a.w + w1 * c.w + w2 * e.w;
      *(float4*)(o + h * 4) = v;
    }
  }
}

// ---------------------------------------------------------------------------
// Kernel 2: fused MLP (2 layers, f32 WMMA 16x16x4) + LayerNorm.
// Block = 256 threads (8 waves), 32 rows/block.
// LDS: X tile 32x516 (padded, conflict-free A-frag reads) + H1 32x260.
// Each wave owns one 16-row x 64-col strip = 4 WMMA tiles.
// ---------------------------------------------------------------------------
#define XSTR 516
#define HSTR 260

__device__ __forceinline__ v8f wmma_f32(v2f a, v2f b, v8f c) {
  return __builtin_amdgcn_wmma_f32_16x16x4_f32(
      false, a, false, b, (short)0, c, false, false);
}

__global__ __launch_bounds__(256) void mlp_ln_kernel(
    const float* __restrict__ nf,    // (B*N, 256) interpolated features
    const float* __restrict__ f2,    // (B*N, 256) feature_2
    const float* __restrict__ W1,    // (512, 256) row-major [k][n]
    const float* __restrict__ b1,    // (256)
    const float* __restrict__ W2,    // (256, 256)
    const float* __restrict__ b2,    // (256)
    const float* __restrict__ gamma, // (256)
    const float* __restrict__ beta,  // (256)
    float* __restrict__ out)         // (B*N, 256)
{
  extern __shared__ float smem[];
  float* X  = smem;               // 32 x XSTR  (concat input, then reused for h2)
  float* H1 = smem + 32 * XSTR;   // 32 x HSTR

  const int tid  = threadIdx.x;
  const int lane = tid & 31;
  const int w    = tid >> 5;
  const int hi   = lane >> 4;     // half-wave select (K offset for A/B frags)
  const int lm   = lane & 15;     // M for A-frag rows, N for B-frag cols
  const size_t r0 = (size_t)blockIdx.x * 32;

  // ---- stage X = [nf | f2] tile: 32 rows x 512 cols, float4 coalesced ----
  #pragma unroll
  for (int i = 0; i < 16; ++i) {
    const int f   = i * 256 + tid;   // float4 slot in [0, 32*128)
    const int row = f >> 7;
    const int c4  = f & 127;
    const float* src = (c4 < 64) ? (nf + (r0 + row) * 256 + c4 * 4)
                                 : (f2 + (r0 + row) * 256 + (c4 - 64) * 4);
    *(float4*)(X + row * XSTR + c4 * 4) = *(const float4*)src;
  }
  __syncthreads();

  const int rt  = w >> 2;        // row tile 0..1  (rows rt*16 .. rt*16+15)
  const int ct0 = (w & 3) * 4;   // first of 4 col tiles (each 16 wide)
  const int c0 = ct0 * 16 + lm, c1 = c0 + 16, c2 = c0 + 32, c3 = c0 + 48;

  // ---- layer 1: (32x512) @ (512x256), K-loop of f32 WMMAs ----
  v8f a0 = {}, a1 = {}, a2 = {}, a3 = {};
  {
    const float* Arow = X + (rt * 16 + lm) * XSTR + 2 * hi;
    for (int k0 = 0; k0 < 512; k0 += 4) {
      const v2f a = *(const v2f*)(Arow + k0);
      const float* wb = W1 + (size_t)(k0 + 2 * hi) * 256;
      v2f f0, f1, f2_, f3;
      f0.x = wb[c0];       f0.y = wb[256 + c0];
      f1.x = wb[c1];       f1.y = wb[256 + c1];
      f2_.x = wb[c2];      f2_.y = wb[256 + c2];
      f3.x = wb[c3];       f3.y = wb[256 + c3];
      a0 = wmma_f32(a, f0, a0);
      a1 = wmma_f32(a, f1, a1);
      a2 = wmma_f32(a, f2_, a2);
      a3 = wmma_f32(a, f3, a3);
    }
  }
  // bias + relu -> H1 (LDS)
  {
    const float bb0 = b1[c0], bb1 = b1[c1], bb2 = b1[c2], bb3 = b1[c3];
    #pragma unroll
    for (int r = 0; r < 8; ++r) {
      float* hrow = H1 + (rt * 16 + r + 8 * hi) * HSTR;
      hrow[c0] = fmaxf(a0[r] + bb0, 0.0f);
      hrow[c1] = fmaxf(a1[r] + bb1, 0.0f);
      hrow[c2] = fmaxf(a2[r] + bb2, 0.0f);
      hrow[c3] = fmaxf(a3[r] + bb3, 0.0f);
    }
  }
  __syncthreads();

  // ---- layer 2: (32x256) @ (256x256) ----
  v8f g0 = {}, g1 = {}, g2 = {}, g3 = {};
  {
    const float* Arow = H1 + (rt * 16 + lm) * HSTR + 2 * hi;
    for (int k0 = 0; k0 < 256; k0 += 4) {
      const v2f a = *(const v2f*)(Arow + k0);
      const float* wb = W2 + (size_t)(k0 + 2 * hi) * 256;
      v2f f0, f1, f2_, f3;
      f0.x = wb[c0];       f0.y = wb[256 + c0];
      f1.x = wb[c1];       f1.y = wb[256 + c1];
      f2_.x = wb[c2];      f2_.y = wb[256 + c2];
      f3.x = wb[c3];       f3.y = wb[256 + c3];
      g0 = wmma_f32(a, f0, g0);
      g1 = wmma_f32(a, f1, g1);
      g2 = wmma_f32(a, f2_, g2);
      g3 = wmma_f32(a, f3, g3);
    }
  }
  // bias + relu -> H2 (reuse X region, stride HSTR)
  float* H2 = X;
  {
    const float bb0 = b2[c0], bb1 = b2[c1], bb2 = b2[c2], bb3 = b2[c3];
    #pragma unroll
    for (int r = 0; r < 8; ++r) {
      float* hrow = H2 + (rt * 16 + r + 8 * hi) * HSTR;
      hrow[c0] = fmaxf(g0[r] + bb0, 0.0f);
      hrow[c1] = fmaxf(g1[r] + bb1, 0.0f);
      hrow[c2] = fmaxf(g2[r] + bb2, 0.0f);
      hrow[c3] = fmaxf(g3[r] + bb3, 0.0f);
    }
  }
  __syncthreads();

  // ---- LayerNorm: 8 lanes per row, shfl_xor reduction ----
  const int row  = tid >> 3;      // 0..31
  const int part = tid & 7;       // 32 channels each
  const float* hrow = H2 + row * HSTR + part * 32;
  float s = 0.0f, ss = 0.0f;
  #pragma unroll
  for (int j = 0; j < 8; ++j) {
    const float4 v = *(const float4*)(hrow + j * 4);
    s  += v.x + v.y + v.z + v.w;
    ss += v.x * v.x + v.y * v.y + v.z * v.z + v.w * v.w;
  }
  #pragma unroll
  for (int off = 1; off < 8; off <<= 1) {
    s  += __shfl_xor(s, off, 32);
    ss += __shfl_xor(ss, off, 32);
  }
  const float mu   = s * (1.0f / 256.0f);
  const float var  = ss * (1.0f / 256.0f) - mu * mu;
  const float rstd = rsqrtf(var + 1e-5f);

  float* orow = out + (r0 + row) * 256 + part * 32;
  const float* gp = gamma + part * 32;
  const float* bp = beta + part * 32;
  #pragma unroll
  for (int j = 0; j < 8; ++j) {
    const float4 v  = *(const float4*)(hrow + j * 4);
    const float4 gv = *(const float4*)(gp + j * 4);
    const float4 bv = *(const float4*)(bp + j * 4);
    float4 o;
    o.x = gv.x * (v.x - mu) * rstd + bv.x;
    o.y = gv.y * (v.y - mu) * rstd + bv.y;
    o.z = gv.z * (v.z - mu) * rstd + bv.z;
    o.w = gv.w * (v.w - mu) * rstd + bv.w;
    *(float4*)(orow + j * 4) = o;
  }
}

// ---------------------------------------------------------------------------
extern "C" void kernel_launch(void* const* d_in, const int* in_sizes, int n_in,
                              void* d_out, int out_size, void* d_ws, size_t ws_size,
                              hipStream_t stream) {
  const float* feature_1 = (const float*)d_in[0];
  const float* feature_2 = (const float*)d_in[1];
  const float* points_1  = (const float*)d_in[2];
  const float* points_2  = (const float*)d_in[3];
  const float* W1        = (const float*)d_in[4];
  const float* b1        = (const float*)d_in[5];
  const float* W2        = (const float*)d_in[6];
  const float* b2        = (const float*)d_in[7];
  const float* gamma     = (const float*)d_in[8];
  const float* beta      = (const float*)d_in[9];
  float* out = (float*)d_out;
  float* nf  = (float*)d_ws;   // (B*N, 256) fp32 = 64 MB scratch

  (void)in_sizes; (void)n_in; (void)out_size; (void)ws_size;

  const int total_rows = Bb * Nn;                    // 65536
  knn_interp_kernel<<<total_rows / 64, 256, 0, stream>>>(
      feature_1, points_1, points_2, nf);

  const size_t smem2 = (size_t)(32 * XSTR + 32 * HSTR) * sizeof(float); // ~97 KB
  mlp_ln_kernel<<<total_rows / 32, 256, smem2, stream>>>(
      nf, feature_2, W1, b1, W2, b2, gamma, beta, out);
}